// GraphConvolutionLayer_25409026523742
// MI455X (gfx1250) — compile-verified
//
#include <hip/hip_runtime.h>

// Problem constants (from reference)
#define B_    8
#define N_    1024
#define HID_  96
#define FEAT_ 32
#define IN_   128
#define OUT_  64
#define R_    4
#define RTOT_ 5

typedef float v2f __attribute__((ext_vector_type(2)));
typedef float v8f __attribute__((ext_vector_type(8)));

// ---------------------------------------------------------------------------
// Kernel 1: supports[b,j,o,r] = x[b,j,:] . W_rel[r,:,o] + b_rel[r,o]
//           self[b,j,o]       = x[b,j,:] . W_self[:,o]  + b_self[o]
// x = concat(hidden[96], node_feats[32]).  One block per (b,j), 64 threads
// (one per output column o).  W matrices (160KB) are L2-resident and read
// coalesced; x row broadcast from LDS.
// Sp layout: Sp[((b*N+j)*64 + o)*4 + r]  -> exactly the B-fragment layout
// needed by the WMMA aggregation kernel (K pair contiguous per lane).
// ---------------------------------------------------------------------------
__global__ __launch_bounds__(64) void rgcn_supports_kernel(
    const float* __restrict__ hidden, const float* __restrict__ nf,
    const float* __restrict__ W_rel,  const float* __restrict__ b_rel,
    const float* __restrict__ W_self, const float* __restrict__ b_self,
    float* __restrict__ Sp, float* __restrict__ C0)
{
    __shared__ float xrow[IN_];
    const int b = blockIdx.y;
    const int j = blockIdx.x;
    const int o = threadIdx.x;              // 0..63

    const size_t row = (size_t)(b * N_ + j);
    for (int idx = threadIdx.x; idx < IN_; idx += 64) {
        xrow[idx] = (idx < HID_) ? hidden[row * HID_ + idx]
                                 : nf[row * FEAT_ + (idx - HID_)];
    }
    __syncthreads();

    float a0 = b_rel[0 * OUT_ + o];
    float a1 = b_rel[1 * OUT_ + o];
    float a2 = b_rel[2 * OUT_ + o];
    float a3 = b_rel[3 * OUT_ + o];
    float as = b_self[o];

    #pragma unroll 4
    for (int i = 0; i < IN_; ++i) {
        const float xv = xrow[i];
        a0 = fmaf(xv, W_rel[(0 * IN_ + i) * OUT_ + o], a0);
        a1 = fmaf(xv, W_rel[(1 * IN_ + i) * OUT_ + o], a1);
        a2 = fmaf(xv, W_rel[(2 * IN_ + i) * OUT_ + o], a2);
        a3 = fmaf(xv, W_rel[(3 * IN_ + i) * OUT_ + o], a3);
        as = fmaf(xv, W_self[i * OUT_ + o], as);
    }

    float4 sv = make_float4(a0, a1, a2, a3);
    *(float4*)(Sp + (row * OUT_ + o) * R_) = sv;    // 16B aligned
    C0[row * OUT_ + o] = as;
}

// ---------------------------------------------------------------------------
// Kernel 2: out[b,i,o] = relu( sum_{j,r} adj[b,i,j,r+1]*Sp[b,j,o,r] + C0 )
// Per batch: 1024 x (K=4096) x 64 GEMM via V_WMMA_F32_16X16X4_F32
// (K=4 per WMMA == the 4 relations of one j).
// Block = 8 waves = one (b, 16-row i-tile).  Wave w covers j in
// [w*128, w*128+128).  Each wave holds 4 C fragments (16x64 tile),
// then cross-wave reduction through LDS, add self-term, ReLU.
//
// A fragment (16x4 f32, ISA layout): lane L (hi=L>>4, lr=L&15):
//   elem0 = adj[b, i0+lr, j, 1+2*hi]  (K=2*hi)
//   elem1 = adj[b, i0+lr, j, 2+2*hi]  (K=2*hi+1)   -> 2 scalar b32 loads,
//   pointer advances 5 floats per j (sequential stream per row).
// B fragment (4x16 f32): lane L: col = 16t+lr,
//   elem0/1 = Sp[b, j, col, 2*hi / 2*hi+1]  -> one aligned 8B load.
// ---------------------------------------------------------------------------
__global__ __launch_bounds__(256) void rgcn_agg_kernel(
    const float* __restrict__ adj, const float* __restrict__ Sp,
    const float* __restrict__ C0,  float* __restrict__ out)
{
    __shared__ float red[8 * 16 * OUT_];    // 8 waves x 16x64 partials = 32KB

    const int b    = blockIdx.y;
    const int i0   = blockIdx.x * 16;
    const int tid  = threadIdx.x;
    const int lane = tid & 31;
    const int w    = tid >> 5;              // wave id 0..7
    const int hi   = lane >> 4;             // 0/1
    const int lr   = lane & 15;

    const int j0 = w * 128;

    // per-lane A stream pointer: adj[b][i0+lr][j0][1+2*hi]
    const float* aPtr = adj
        + ((size_t)(b * N_ + i0 + lr) * N_ + j0) * RTOT_ + 1 + 2 * hi;
    // per-lane B stream pointer: Sp[b][j0][lr][2*hi]
    const float* bPtr = Sp
        + (size_t)(b * N_ + j0) * (OUT_ * R_) + lr * R_ + 2 * hi;

    v8f c0 = {}, c1 = {}, c2 = {}, c3 = {};

    #pragma unroll 4
    for (int j = 0; j < 128; ++j) {
        v2f a;
        a.x = aPtr[0];
        a.y = aPtr[1];
        v2f bv0 = *(const v2f*)(bPtr + 0 * 16 * R_);
        v2f bv1 = *(const v2f*)(bPtr + 1 * 16 * R_);
        v2f bv2 = *(const v2f*)(bPtr + 2 * 16 * R_);
        v2f bv3 = *(const v2f*)(bPtr + 3 * 16 * R_);
        c0 = __builtin_amdgcn_wmma_f32_16x16x4_f32(false, a, false, bv0,
                                                   (short)0, c0, false, false);
        c1 = __builtin_amdgcn_wmma_f32_16x16x4_f32(false, a, false, bv1,
                                                   (short)0, c1, false, false);
        c2 = __builtin_amdgcn_wmma_f32_16x16x4_f32(false, a, false, bv2,
                                                   (short)0, c2, false, false);
        c3 = __builtin_amdgcn_wmma_f32_16x16x4_f32(false, a, false, bv3,
                                                   (short)0, c3, false, false);
        aPtr += RTOT_;
        bPtr += OUT_ * R_;
    }

    // spill partial C fragments: red[w][m][n]
    // C layout: VGPR v, lane L: M = v + 8*hi, N = 16t + lr
    float* rw = red + w * (16 * OUT_);
    #pragma unroll
    for (int v = 0; v < 8; ++v) {
        const int m = v + 8 * hi;
        rw[m * OUT_ + 0 * 16 + lr] = c0[v];
        rw[m * OUT_ + 1 * 16 + lr] = c1[v];
        rw[m * OUT_ + 2 * 16 + lr] = c2[v];
        rw[m * OUT_ + 3 * 16 + lr] = c3[v];
    }
    __syncthreads();

    // cross-wave reduction: 256 threads x 4 contiguous outputs each
    const size_t obase = (size_t)(b * N_ + i0) * OUT_;
    float4 s = make_float4(0.f, 0.f, 0.f, 0.f);
    #pragma unroll
    for (int ww = 0; ww < 8; ++ww) {
        const float4 p = *(const float4*)(red + ww * (16 * OUT_) + tid * 4);
        s.x += p.x; s.y += p.y; s.z += p.z; s.w += p.w;
    }
    const float4 cc = *(const float4*)(C0 + obase + tid * 4);
    float4 r;
    r.x = fmaxf(s.x + cc.x, 0.f);
    r.y = fmaxf(s.y + cc.y, 0.f);
    r.z = fmaxf(s.z + cc.z, 0.f);
    r.w = fmaxf(s.w + cc.w, 0.f);
    *(float4*)(out + obase + tid * 4) = r;
}

// ---------------------------------------------------------------------------
extern "C" void kernel_launch(void* const* d_in, const int* in_sizes, int n_in,
                              void* d_out, int out_size, void* d_ws, size_t ws_size,
                              hipStream_t stream) {
    const float* adj    = (const float*)d_in[0];
    const float* hidden = (const float*)d_in[1];
    const float* nf     = (const float*)d_in[2];
    const float* W_rel  = (const float*)d_in[3];
    const float* b_rel  = (const float*)d_in[4];
    const float* W_self = (const float*)d_in[5];
    const float* b_self = (const float*)d_in[6];
    float* out = (float*)d_out;

    float* Sp = (float*)d_ws;                              // 8 MB supports
    float* C0 = Sp + (size_t)B_ * N_ * OUT_ * R_;          // 2 MB self-term

    rgcn_supports_kernel<<<dim3(N_, B_), 64, 0, stream>>>(
        hidden, nf, W_rel, b_rel, W_self, b_self, Sp, C0);

    rgcn_agg_kernel<<<dim3(N_ / 16, B_), 256, 0, stream>>>(
        adj, Sp, C0, out);
}